// LSSLKernel_10393820856910
// MI455X (gfx1250) — compile-verified
//
#include <hip/hip_runtime.h>
#include <hip/hip_bf16.h>
#include <math.h>

// ---------------------------------------------------------------------------
// S4 NPLR Cauchy kernel + Woodbury + length-8192 inverse DFT for MI455X
// (gfx1250, wave32).
//
//   d = 256 channels, N = 64 states, L = 8192 sequence length.
//
// Stage 1 (cauchy_kernel): K_hat[d,l] via Cauchy resolvent + rank-1 Woodbury.
// Stage 2 (idft8192_kernel): K[d,:] = Re(ifft(K_hat[d,:])) using the
//   four-step decomposition 8192 = 64 x 128:
//       X[a,b] = K_hat[a + 64 b]            (a<64, b<128)
//       Y[a,k0] = sum_b X[a,b] e^{+2πi b k0/128}        (GEMM vs F128)
//       Z[a,k0] = Y[a,k0] e^{+2πi a k0/8192}            (twiddle)
//       K[k0+128 k1] = (1/L) Re sum_a e^{+2πi a k1/64} Z[a,k0]  (GEMM vs F64)
//   Both GEMMs run on V_WMMA_F32_16X16X4_F32 (fp32-exact).
//
// Transcendentals: only sincospif (exact binary-fraction arguments, no
// Payne-Hanek path), and only during LUT setup / per-thread z.  The 8192-pt
// twiddle is the product of a coarse (2πi j/128) and fine (2πi j/8192) LUT.
//
// Register-pressure design: each GEMM is split into two half-passes over the
// k0 tiles so only 8 v8f accumulators (64 VGPRs) are live at once.  GEMM1
// writes a separate LDS plane sY so half-pass 2 still sees untouched X data.
//
// WMMA operand layouts (CDNA5 ISA 7.12.2):
//   A 16x4 f32 : lane -> M = lane%16 ; VGPR r -> K = 2*(lane/16)+r
//   B 4x16 f32 : lane -> N = lane%16 ; VGPR r -> K = 2*(lane/16)+r   (mirror)
//   C/D 16x16  : lane -> N = lane%16 ; VGPR r -> M = 8*(lane/16)+r
//
// Workspace: K_hat as float2[256*8192] = 16 MB (L2-resident, 192 MB L2).
// LDS per block: 64 KB (X) + 64 KB (Y) + 1.5 KB (LUTs) of 320 KB/WGP.
// ---------------------------------------------------------------------------

#define DM   256
#define NS   64
#define LL   8192
#define EPSF 1.1920929e-7f

typedef float v2f __attribute__((ext_vector_type(2)));
typedef float v8f __attribute__((ext_vector_type(8)));

#define WMMA_F32(a, b, c) \
  __builtin_amdgcn_wmma_f32_16x16x4_f32(false, (a), false, (b), (short)0, (c), false, false)

// ------------------------------ Stage 1 ------------------------------------
__global__ void cauchy_kernel(const float* __restrict__ Lr,
                              const float* __restrict__ Li,
                              const float* __restrict__ Pr,
                              const float* __restrict__ Pi,
                              const float* __restrict__ Br,
                              const float* __restrict__ Bi,
                              const float* __restrict__ log_dt,
                              float2* __restrict__ KH) {
  const int d = blockIdx.x;
  const int l = blockIdx.y * blockDim.x + threadIdx.x;

  __shared__ float sLr[NS], sLi[NS], sPr[NS], sPi[NS], sBr[NS], sBi[NS];
  const int t = threadIdx.x;
  if (t < NS) {
    float x = Lr[d * NS + t];
    float sp = (x > 20.0f) ? x : log1pf(expf(x));   // softplus
    sLr[t] = -sp;                                   // Lam_re = -softplus
    sLi[t] = Li[d * NS + t];
    sPr[t] = Pr[d * NS + t];
    sPi[t] = Pi[d * NS + t];
    sBr[t] = Br[d * NS + t];
    sBi[t] = Bi[d * NS + t];
  }
  __syncthreads();

  const float dt = expf(log_dt[d]);
  const float s2 = 2.0f / dt;

  // z = exp(-2πi l / L) = exp(-iπ * l/4096); l/4096 is an exact fraction
  float zs, zc;
  sincospif(-(float)l * (1.0f / 4096.0f), &zs, &zc);
  float dre = 1.0f + zc;      // denom_z = 1 + z
  float dim = zs;
  if (sqrtf(dre * dre + dim * dim) < EPSF) { dre = EPSF; dim = 0.0f; }
  const float invd = 1.0f / (dre * dre + dim * dim);

  // g = (2/dt) * (1 - z) / denom_z
  const float nre = 1.0f - zc, nim = -zs;
  const float gre = s2 * (nre * dre + nim * dim) * invd;
  const float gim = s2 * (nim * dre - nre * dim) * invd;

  float PRBr = 0.f, PRBi = 0.f, PRPr = 0.f, PRPi = 0.f;
  float BRBr = 0.f, BRBi = 0.f, BRPr = 0.f, BRPi = 0.f;

#pragma unroll 8
  for (int n = 0; n < NS; ++n) {
    const float ar = gre - sLr[n];
    const float ai = gim - sLi[n];
    const float im = 1.0f / (ar * ar + ai * ai);
    const float rre = ar * im;
    const float rim = -ai * im;
    const float pre = sPr[n], pim = sPi[n];
    const float bre = sBr[n], bim = sBi[n];
    // t1 = conj(P) * R ; t2 = conj(B) * R
    const float t1r = pre * rre + pim * rim;
    const float t1i = pre * rim - pim * rre;
    const float t2r = bre * rre + bim * rim;
    const float t2i = bre * rim - bim * rre;
    PRBr += t1r * bre - t1i * bim;  PRBi += t1r * bim + t1i * bre;
    PRPr += t1r * pre - t1i * pim;  PRPi += t1r * pim + t1i * pre;
    BRBr += t2r * bre - t2i * bim;  BRBi += t2r * bim + t2i * bre;
    BRPr += t2r * pre - t2i * pim;  BRPi += t2r * pim + t2i * pre;
  }

  // q = PRB / (1 + PRP)
  const float wr = 1.0f + PRPr, wi = PRPi;
  const float iw = 1.0f / (wr * wr + wi * wi);
  const float qr = (PRBr * wr + PRBi * wi) * iw;
  const float qi = (PRBi * wr - PRBr * wi) * iw;
  // w = BRB - BRP * q
  const float wre = BRBr - (BRPr * qr - BRPi * qi);
  const float wim = BRBi - (BRPr * qi + BRPi * qr);
  // K_hat = (2 / denom_z) * w = 2 * w * conj(denom_z) * invd
  const float kre = 2.0f * invd * (wre * dre + wim * dim);
  const float kim = 2.0f * invd * (wim * dre - wre * dim);
  KH[(size_t)d * LL + l] = make_float2(kre, kim);
}

// ------------------------------ Stage 2 ------------------------------------
// One workgroup (128 threads = 4 waves) per channel d; each wave owns a
// 16-row stripe of the 64x128 tile.
__global__ void __launch_bounds__(128, 1)
idft8192_kernel(const float2* __restrict__ KH, float* __restrict__ out) {
  extern __shared__ char smem[];
  float2* sX  = (float2*)smem;                              // [a*128 + b]
  float2* sY  = (float2*)(smem + 64 * 128 * sizeof(float2));// [a*128 + k0]
  float*  tabc = (float*)(smem + 2 * 64 * 128 * sizeof(float2));
  float*  tabs = tabc + 128;   // coarse LUT: exp(+2πi j/128)
  float*  tfc  = tabs + 128;   // fine LUT:   exp(+2πi j/8192), j < 64
  float*  tfs  = tfc + 64;

  const int d    = blockIdx.x;
  const int tid  = threadIdx.x;
  const int lane = tid & 31;
  const int wave = tid >> 5;
  const int m    = lane & 15;   // row (A) / column (B,C/D) within tile
  const int h    = lane >> 4;   // lane half

  if (tid < 128) {              // exp(+2πi tid/128) = exp(+iπ tid/64)
    float s, c;
    sincospif((float)tid * (1.0f / 64.0f), &s, &c);
    tabc[tid] = c;
    tabs[tid] = s;
  }
  if (tid < 64) {               // exp(+2πi tid/8192) = exp(+iπ tid/4096)
    float s, c;
    sincospif((float)tid * (1.0f / 4096.0f), &s, &c);
    tfc[tid] = c;
    tfs[tid] = s;
  }
  // load X[a][b] = K_hat[a + 64 b]
  const float2* kh = KH + (size_t)d * LL;
  for (int idx = tid; idx < 64 * 128; idx += 128) {
    const int a = idx >> 7, b = idx & 127;
    sX[idx] = kh[a + (b << 6)];
  }
  __syncthreads();

  const int arow  = wave * 16 + m;
  const int k1row = wave * 16 + m;
  const float4* sX4 = (const float4*)sX;   // two adjacent complex elems

  // ---- GEMM 1: Y[a,k0] = sum_b X[a,b] * exp(2πi b k0/128); Z = twiddled ----
  for (int half = 0; half < 2; ++half) {
    v8f yre[4], yim[4];
#pragma unroll
    for (int i = 0; i < 4; ++i) {
      yre[i] = (v8f){0, 0, 0, 0, 0, 0, 0, 0};
      yim[i] = (v8f){0, 0, 0, 0, 0, 0, 0, 0};
    }
    for (int kt = 0; kt < 32; ++kt) {             // K dim: b, 4 per step
      const int b0 = kt * 4 + 2 * h;
      const float4 q = sX4[(arow * 128 + b0) >> 1];   // X[arow, b0..b0+1]
      const v2f are = {q.x, q.z};
      const v2f aim = {q.y, q.w};
      const v2f nim = {-q.y, -q.w};
#pragma unroll
      for (int j = 0; j < 4; ++j) {               // N dim: k0 tiles
        const int k0 = (half * 4 + j) * 16 + m;
        const int i0 = (b0 * k0) & 127;
        const int i1 = ((b0 + 1) * k0) & 127;
        const v2f bc = {tabc[i0], tabc[i1]};
        const v2f bs = {tabs[i0], tabs[i1]};
        yre[j] = WMMA_F32(are, bc, yre[j]);
        yre[j] = WMMA_F32(nim, bs, yre[j]);       // - Xim * S
        yim[j] = WMMA_F32(are, bs, yim[j]);
        yim[j] = WMMA_F32(aim, bc, yim[j]);
      }
    }
    // twiddle Z = Y * exp(2πi a k0/8192) into sY (own rows only).
    // exp(2πi t/8192) = exp(2πi (t>>6)/128) * exp(2πi (t&63)/8192), t = a*k0.
#pragma unroll
    for (int j = 0; j < 4; ++j) {
      const int k0 = (half * 4 + j) * 16 + m;
#pragma unroll
      for (int r = 0; r < 8; ++r) {
        const int a = wave * 16 + 8 * h + r;      // C/D layout row
        const int tt = a * k0;                    // < 8192
        const int t1 = tt >> 6, t0 = tt & 63;
        const float cc = tabc[t1], cs = tabs[t1];
        const float fc = tfc[t0],  fs = tfs[t0];
        const float c = cc * fc - cs * fs;
        const float s = cc * fs + cs * fc;
        const float yr = yre[j][r], yi = yim[j][r];
        sY[a * 128 + k0] = make_float2(yr * c - yi * s, yr * s + yi * c);
      }
    }
  }
  __syncthreads();

  // ---- GEMM 2: K[k0+128 k1] = Re sum_a exp(2πi a k1/64) Z[a,k0] ----
  float* o = out + (size_t)d * LL;
  for (int half = 0; half < 2; ++half) {
    v8f acc[4];
#pragma unroll
    for (int i = 0; i < 4; ++i) acc[i] = (v8f){0, 0, 0, 0, 0, 0, 0, 0};
    for (int kt = 0; kt < 16; ++kt) {             // K dim: a, 4 per step
      const int a0 = kt * 4 + 2 * h;
      const int i0 = (2 * a0 * k1row) & 127;      // exp(2πi a k1/64) via LUT
      const int i1 = (2 * (a0 + 1) * k1row) & 127;
      const v2f ac  = {tabc[i0], tabc[i1]};
      const v2f nas = {-tabs[i0], -tabs[i1]};
#pragma unroll
      for (int j = 0; j < 4; ++j) {
        const int k0 = (half * 4 + j) * 16 + m;
        const float2 z0 = sY[a0 * 128 + k0];
        const float2 z1 = sY[(a0 + 1) * 128 + k0];
        const v2f bre = {z0.x, z1.x};
        const v2f bim = {z0.y, z1.y};
        acc[j] = WMMA_F32(ac, bre, acc[j]);       //  C64 @ Zre
        acc[j] = WMMA_F32(nas, bim, acc[j]);      // -S64 @ Zim
      }
    }
#pragma unroll
    for (int j = 0; j < 4; ++j) {
      const int k0 = (half * 4 + j) * 16 + m;
#pragma unroll
      for (int r = 0; r < 8; ++r) {
        const int k1 = wave * 16 + 8 * h + r;
        o[k0 + 128 * k1] = acc[j][r] * (1.0f / (float)LL);
      }
    }
  }
}

// ------------------------------ Stage 3 ------------------------------------
__global__ void zero_tail_kernel(float* __restrict__ dtail) {
  dtail[threadIdx.x] = 0.0f;   // D output buffer is all zeros in the reference
}

// ---------------------------------------------------------------------------
extern "C" void kernel_launch(void* const* d_in, const int* in_sizes, int n_in,
                              void* d_out, int out_size, void* d_ws, size_t ws_size,
                              hipStream_t stream) {
  (void)in_sizes; (void)n_in; (void)out_size; (void)ws_size;
  const float* Lr  = (const float*)d_in[0];
  const float* Li  = (const float*)d_in[1];
  const float* Pr  = (const float*)d_in[2];
  const float* Pi  = (const float*)d_in[3];
  const float* Br  = (const float*)d_in[4];
  const float* Bi  = (const float*)d_in[5];
  const float* ldt = (const float*)d_in[6];
  float* out = (float*)d_out;

  float2* KH = (float2*)d_ws;   // 256*8192 float2 = 16 MB scratch

  cauchy_kernel<<<dim3(DM, LL / 256), 256, 0, stream>>>(Lr, Li, Pr, Pi, Br, Bi,
                                                        ldt, KH);

  const size_t shmem = 2 * 64 * 128 * sizeof(float2)   // sX + sY
                     + 2 * 128 * sizeof(float)         // coarse LUT
                     + 2 * 64 * sizeof(float);         // fine LUT
  (void)hipFuncSetAttribute((const void*)idft8192_kernel,
                            hipFuncAttributeMaxDynamicSharedMemorySize,
                            (int)shmem);
  idft8192_kernel<<<DM, 128, shmem, stream>>>(KH, out);

  zero_tail_kernel<<<1, DM, 0, stream>>>(out + (size_t)DM * LL);
}